// fivevariate_poly_23484881174592
// MI455X (gfx1250) — compile-verified
//
#include <hip/hip_runtime.h>

typedef __attribute__((ext_vector_type(2))) float v2f;
typedef __attribute__((ext_vector_type(4))) float v4f;
typedef __attribute__((ext_vector_type(8))) float v8f;

#define DEG 10

// Each wave computes a 16-row tile:
//   out[m] = sum_{i,d} Wdir[i,d] p_i[m,d]  +  sum_k p_{i_k}[m]^T Wk p_{j_k}[m]
// Cross terms via fp32 WMMA (the only full-precision matrix op on CDNA5):
//   Yk = P_j(16x10) x Wk^T(10x10) as 3x v_wmma_f32_16x16x4_f32 (K padded to 12, N to 16),
// then an elementwise contraction with p_i held in C-matrix layout (staged via LDS).
// B operands come from a zero-padded LDS weight image -> branch-free ds_load_b64.
__launch_bounds__(256, 1)
__global__ void poly5_wmma_kernel(const float* __restrict__ x,
                                  const float* __restrict__ W,
                                  float* __restrict__ out, int Brows)
{
    const int lane = threadIdx.x & 31;
    const int wave = threadIdx.x >> 5;
    const int n    = lane & 15;   // column index (N / degree) for B,C layouts; row for A
    const int h    = lane >> 4;   // K-half for A/B layouts, row-half for C layout
    const int m    = n;           // A-matrix row handled by this lane

    const int tile    = blockIdx.x * 8 + wave;
    const int rowBase = tile * 16;

    // ---- LDS: zero-padded weight image (8 KB) + per-wave power staging (40 KB) ----
    __shared__ float WL[2048];                  // [k][n:16][dj:12] cross (1920) | direct [5][16] @1920
    __shared__ float lds[8 * 5 * 16 * 16];      // 8 waves x 5 vars x 16 rows x 16 degs

    // Cooperative branch-light fill of the padded weight image (once per block).
    for (int idx = threadIdx.x; idx < 2000; idx += 256) {
        bool ok; int src;
        if (idx < 1920) {
            const int k  = idx / 192;
            const int nn = (idx % 192) / 12;
            const int dj = idx % 12;
            ok  = (nn < DEG) && (dj < DEG);
            src = 5 * DEG + k * 100 + nn * DEG + dj;
        } else {
            const int t  = idx - 1920;
            const int nn = t & 15;
            ok  = (nn < DEG);
            src = (t >> 4) * DEG + nn;
        }
        const float tv = W[ok ? src : 0];       // clamped load + select, no divergence
        WL[idx] = ok ? tv : 0.0f;
    }

    // ---- per-lane powers of its A-row (row = rowBase + m); halves duplicate ----
    int row = rowBase + m;
    if (row >= Brows) row = Brows - 1;          // clamp (B is a multiple of 16 anyway)
    float pw[5][DEG];
    #pragma unroll
    for (int v = 0; v < 5; ++v) {
        const float xv = x[row * 5 + v];
        pw[v][0] = xv;
        #pragma unroll
        for (int d = 1; d < DEG; ++d) pw[v][d] = pw[v][d - 1] * xv;
    }

    // ---- stage powers to LDS [var][row][deg padded to 16] as b128 stores ----
    float* myLds = lds + wave * (5 * 16 * 16);
    #pragma unroll
    for (int v = 0; v < 5; ++v) {
        const bool mine = (h == 0) ? (v < 3) : (v >= 3);   // split writes across halves
        if (mine) {
            v4f q0 = {pw[v][0], pw[v][1], pw[v][2], pw[v][3]};
            v4f q1 = {pw[v][4], pw[v][5], pw[v][6], pw[v][7]};
            v4f q2 = {pw[v][8], pw[v][9], 0.0f, 0.0f};
            v4f q3 = {0.0f, 0.0f, 0.0f, 0.0f};
            v4f* dst = (v4f*)&myLds[v * 256 + m * 16];
            dst[0] = q0; dst[1] = q1; dst[2] = q2; dst[3] = q3;
        }
    }
    __syncthreads();

    float pwT[5][8];                             // pwT[v][r] = p_v[row r+8h, deg n]
    #pragma unroll
    for (int v = 0; v < 5; ++v)
        #pragma unroll
        for (int r = 0; r < 8; ++r)
            pwT[v][r] = myLds[v * 256 + (r + 8 * h) * 16 + n];

    // ---- A operands (shared by pairs with same j): lane holds degs {4c+2h, 4c+2h+1} ----
    v2f aop[4][3];
    #pragma unroll
    for (int j = 1; j < 5; ++j) {
        #pragma unroll
        for (int c = 0; c < 3; ++c) {
            const int d0 = 4 * c;
            const float a0l = pw[j][d0 + 0];
            const float a1l = pw[j][d0 + 1];
            const float a0h = (d0 + 2 < DEG) ? pw[j][d0 + 2] : 0.0f;
            const float a1h = (d0 + 3 < DEG) ? pw[j][d0 + 3] : 0.0f;
            aop[j - 1][c].x = h ? a0h : a0l;
            aop[j - 1][c].y = h ? a1h : a1l;
        }
    }

    const int PI[10] = {0, 0, 0, 0, 1, 1, 1, 2, 2, 3};
    const int PJ[10] = {1, 2, 3, 4, 2, 3, 4, 3, 4, 4};

    float contrib[8] = {0.f, 0.f, 0.f, 0.f, 0.f, 0.f, 0.f, 0.f};
    const int bbase = n * 12 + 2 * h;            // lane-constant part of B offsets

    #pragma unroll
    for (int k = 0; k < 10; ++k) {
        const int pi = PI[k], pj = PJ[k];
        v8f acc = {0.f, 0.f, 0.f, 0.f, 0.f, 0.f, 0.f, 0.f};
        #pragma unroll
        for (int c = 0; c < 3; ++c) {
            // B[kk,n] = Wk[di=n, dj=4c+kk]; zero padding already baked into WL.
            // Offset is even -> 8-byte aligned ds_load_b64.
            const v2f b = *(const v2f*)&WL[k * 192 + bbase + 4 * c];
            acc = __builtin_amdgcn_wmma_f32_16x16x4_f32(
                false, aop[pj - 1][c], false, b, (short)0, acc, false, false);
        }
        // contraction with left power vector, exactly in C-matrix layout
        #pragma unroll
        for (int r = 0; r < 8; ++r)
            contrib[r] = fmaf(pwT[pi][r], acc[r], contrib[r]);
    }

    // ---- direct (non-cross) terms, also in C-layout ----
    #pragma unroll
    for (int v = 0; v < 5; ++v) {
        const float wd = WL[1920 + v * 16 + n];
        #pragma unroll
        for (int r = 0; r < 8; ++r)
            contrib[r] = fmaf(pwT[v][r], wd, contrib[r]);
    }

    // ---- reduce over the 16 columns (per half-wave) and store one scalar per row ----
    #pragma unroll
    for (int r = 0; r < 8; ++r) {
        float s = contrib[r];
        s += __shfl_xor(s, 1, 32);
        s += __shfl_xor(s, 2, 32);
        s += __shfl_xor(s, 4, 32);
        s += __shfl_xor(s, 8, 32);
        const int orow = rowBase + 8 * h + r;
        if (n == r && orow < Brows) out[orow] = s;
    }
}

extern "C" void kernel_launch(void* const* d_in, const int* in_sizes, int n_in,
                              void* d_out, int out_size, void* d_ws, size_t ws_size,
                              hipStream_t stream) {
    const float* x  = (const float*)d_in[0];    // [B,5] fp32
    const float* W  = (const float*)d_in[1];    // [1,1050] fp32
    float* out      = (float*)d_out;            // [B,1] fp32
    const int Brows = in_sizes[0] / 5;
    const int tiles  = (Brows + 15) / 16;
    const int blocks = (tiles + 7) / 8;         // 8 waves (16-row tiles) per 256-thread block
    poly5_wmma_kernel<<<blocks, 256, 0, stream>>>(x, W, out, Brows);
}